// BlockB_12137577578731
// MI455X (gfx1250) — compile-verified
//
#include <hip/hip_runtime.h>

// ---------------------------------------------------------------------------
// Grouped 1D conv (ws=1,2,3; F=8 filters/channel) + global max/mean pooling.
// One wave32 per (b,d) pair.
//   * ws=2/3 MAX pooling via V_WMMA_F32_16X16X4_F32 with
//       A (16x4)  = 16 positions x 4 taps of x   (from LDS, per tile)
//       B (4x16)  = 4 taps x [ws2 f0..7 | ws3 f0..7]  (built once)
//     D layout puts FILTER on the lane axis and POSITION on the register
//     axis, so max-pooling reduces in-register (v_max3 trees) with a single
//     cross-half ds_swizzle at the end; outputs store per-lane, coalesced.
//   * ALL mean pooling is analytic: mean_l y = (sum_k w_k * S_k)/L' + b,
//     S_k = (total sum of x) - edge elements  -> zero per-position work.
//   * ws=1 max is analytic from (max x, min x) since it's a scalar scale.
// Memory-optimal (x row read once, 512B b128/lane via LDS; 192B out/wave).
// ---------------------------------------------------------------------------

typedef __attribute__((ext_vector_type(2))) float v2f;
typedef __attribute__((ext_vector_type(8))) float v8f;

#define BATCH 128
#define DCH   300
#define LEN   128
#define NF    8
#define WPB   8          // waves per block

#define DPP_XOR1  0xB1   // quad_perm(1,0,3,2)
#define DPP_XOR2  0x4E   // quad_perm(2,3,0,1)
#define DPP_HMIRR 0x141  // row_half_mirror (combines 4-groups within 8)
#define DPP_MIRR  0x140  // row_mirror      (combines 8-groups within 16)

template <int CTRL>
__device__ __forceinline__ float dpp_bfly(float v) {
    int i = __builtin_bit_cast(int, v);
    int t = __builtin_amdgcn_update_dpp(i, i, CTRL, 0xF, 0xF, false);
    return __builtin_bit_cast(float, t);
}
__device__ __forceinline__ float row16_max(float v) {
    v = fmaxf(v, dpp_bfly<DPP_XOR1>(v));
    v = fmaxf(v, dpp_bfly<DPP_XOR2>(v));
    v = fmaxf(v, dpp_bfly<DPP_HMIRR>(v));
    v = fmaxf(v, dpp_bfly<DPP_MIRR>(v));
    return v;
}
__device__ __forceinline__ float row16_min(float v) {
    v = fminf(v, dpp_bfly<DPP_XOR1>(v));
    v = fminf(v, dpp_bfly<DPP_XOR2>(v));
    v = fminf(v, dpp_bfly<DPP_HMIRR>(v));
    v = fminf(v, dpp_bfly<DPP_MIRR>(v));
    return v;
}
__device__ __forceinline__ float row16_sum(float v) {
    v += dpp_bfly<DPP_XOR1>(v);
    v += dpp_bfly<DPP_XOR2>(v);
    v += dpp_bfly<DPP_HMIRR>(v);
    v += dpp_bfly<DPP_MIRR>(v);
    return v;
}
// cross-half (lane ^ 16) exchange: single ds_swizzle (and=0x1f, or=0, xor=0x10)
__device__ __forceinline__ float swz_xor16(float v) {
    int t = __builtin_amdgcn_ds_swizzle(__builtin_bit_cast(int, v), 0x401F);
    return __builtin_bit_cast(float, t);
}

__global__ __launch_bounds__(256) void conv_pool_wmma(
    const float* __restrict__ x,                                   // [B, D, L]
    const float* __restrict__ W1, const float* __restrict__ b1,
    const float* __restrict__ W2, const float* __restrict__ b2,
    const float* __restrict__ W3, const float* __restrict__ b3,
    float* __restrict__ out)                                       // [B, 2, 3, D, F]
{
    __shared__ float xs[WPB][132];               // 128 + 4 zero pad (x[l+k], k<=3)

    const int lane = threadIdx.x & 31;
    const int wv   = threadIdx.x >> 5;
    const int pair = blockIdx.x * WPB + wv;      // exact fit: 4800*8 = 38400
    const int b    = pair / DCH;
    const int d    = pair - b * DCH;

    // ---- stage x[b,d,:] into LDS: one coalesced 16B load per lane -----------
    const float4 xv = *(const float4*)(x + (size_t)pair * LEN + lane * 4);
    *(float4*)(&xs[wv][lane * 4]) = xv;
    if (lane < 4) xs[wv][LEN + lane] = 0.0f;

    // ---- wave stats: max/min/sum of x (DPP rows + one cross-half swizzle) ---
    float mx = fmaxf(fmaxf(xv.x, xv.y), fmaxf(xv.z, xv.w));
    float mn = fminf(fminf(xv.x, xv.y), fminf(xv.z, xv.w));
    float sm = (xv.x + xv.y) + (xv.z + xv.w);
    mx = row16_max(mx);  mx = fmaxf(mx, swz_xor16(mx));
    mn = row16_min(mn);  mn = fminf(mn, swz_xor16(mn));
    sm = row16_sum(sm);  sm += swz_xor16(sm);

    // ---- B matrix (4x16 f32): N=0..7 ws2 filters, N=8..15 ws3 filters -------
    // Layout: lanes 0-15 hold K={0,1} in VGPR{0,1}; lanes 16-31 hold K={2,3}.
    v2f Bw;
    {
        const int  nn = lane & 15;
        const bool hi = lane >= 16;
        float w0 = 0.0f, w1 = 0.0f;
        if (nn < 8) {                            // ws=2 filter nn: taps 0,1
            if (!hi) { w0 = W2[(d * NF + nn) * 2 + 0]; w1 = W2[(d * NF + nn) * 2 + 1]; }
        } else {                                 // ws=3 filter nn-8: taps 0,1,2
            const int f = nn - 8;
            if (!hi) { w0 = W3[(d * NF + f) * 3 + 0]; w1 = W3[(d * NF + f) * 3 + 1]; }
            else     { w0 = W3[(d * NF + f) * 3 + 2]; }
        }
        Bw[0] = w0; Bw[1] = w1;
    }

    __syncthreads();                             // LDS row visible to whole wave

    // ---- 8 WMMA tiles; per-lane in-register max trees -----------------------
    // D: VGPR j, lane l<16 -> (pos 16t+j,   filter l)
    //            lane l>=16 -> (pos 16t+j+8, filter l-16)
    const int m    = lane & 15;                  // position row within A
    const int kofs = (lane < 16) ? 0 : 2;        // A: K={0,1} lo / K={2,3} hi half

    float acc = -__builtin_inff();
    #pragma unroll
    for (int t = 0; t < 8; ++t) {
        v2f Ap;
        const int off = t * 16 + m + kofs;
        Ap[0] = xs[wv][off];
        Ap[1] = xs[wv][off + 1];
        v8f C = {};
        C = __builtin_amdgcn_wmma_f32_16x16x4_f32(false, Ap, false, Bw,
                                                  (short)0, C, false, false);
        float c6 = C[6], c7 = C[7];
        if (t == 7) {                            // ragged tail, hi half only:
            c6 = (lane < 24) ? c6 : -__builtin_inff();  // l=126 invalid for ws3
            c7 = (lane < 16) ? c7 : -__builtin_inff();  // l=127 invalid for both
        }
        const float r = fmaxf(fmaxf(fmaxf(C[0], C[1]), fmaxf(C[2], C[3])),
                              fmaxf(fmaxf(C[4], C[5]), fmaxf(c6, c7)));
        acc = fmaxf(acc, r);
    }
    // combine the two position-halves of each filter: one swizzle + one max
    acc = fmaxf(acc, swz_xor16(acc));            // lane n (&15) = filter n max

    // ---- analytic means: S_k = sum - edges ----------------------------------
    const float x0   = xs[wv][0],   x1   = xs[wv][1];
    const float x126 = xs[wv][126], x127 = xs[wv][127];
    const float S0_2 = sm - x127,        S1_2 = sm - x0;            // ws2, L'=127
    const float S0_3 = sm - x126 - x127, S1_3 = sm - x0 - x127,     // ws3, L'=126
                S2_3 = sm - x0 - x1;
    const float w3t2 = swz_xor16(Bw[0]);         // lanes 8..15 fetch ws3 tap k=2

    // ---- outputs: [b, pool(2), ws(3), d, f] ; all stores per-lane/coalesced -
    const size_t ob   = (size_t)b * (2 * 3 * DCH * NF);
    const size_t strP = (size_t)3 * DCH * NF;    // pool-axis stride
    const size_t strW = (size_t)DCH * NF;        // ws-axis stride
    float* __restrict__ omax  = out + ob;
    float* __restrict__ omean = out + ob + strP;

    if (lane < 8) {                              // f = lane
        // ws=1: analytic max & mean
        const float w  = W1[d * NF + lane];
        const float bb = b1[d * NF + lane];
        omax [0 * strW + d * NF + lane] = (w >= 0.0f ? w * mx : w * mn) + bb;
        omean[0 * strW + d * NF + lane] = w * (sm * (1.0f / 128.0f)) + bb;
        // ws=2: taps live in this lane's B regs
        const float bias2 = b2[d * NF + lane];
        omax [1 * strW + d * NF + lane] = acc + bias2;
        omean[1 * strW + d * NF + lane] =
            (Bw[0] * S0_2 + Bw[1] * S1_2) * (1.0f / 127.0f) + bias2;
    } else if (lane < 16) {                      // f = lane - 8 : ws=3
        const int f = lane - 8;
        const float bias3 = b3[d * NF + f];
        omax [2 * strW + d * NF + f] = acc + bias3;
        omean[2 * strW + d * NF + f] =
            (Bw[0] * S0_3 + Bw[1] * S1_3 + w3t2 * S2_3) * (1.0f / 126.0f) + bias3;
    }
}

extern "C" void kernel_launch(void* const* d_in, const int* in_sizes, int n_in,
                              void* d_out, int out_size, void* d_ws, size_t ws_size,
                              hipStream_t stream) {
    const float* x  = (const float*)d_in[0];
    const float* W1 = (const float*)d_in[1];
    const float* b1 = (const float*)d_in[2];
    const float* W2 = (const float*)d_in[3];
    const float* b2 = (const float*)d_in[4];
    const float* W3 = (const float*)d_in[5];
    const float* b3 = (const float*)d_in[6];
    float* out = (float*)d_out;

    const int pairs  = BATCH * DCH;              // 38400 waves
    const int blocks = pairs / WPB;              // 4800 (exact)
    conv_pool_wmma<<<blocks, WPB * 32, 0, stream>>>(x, W1, b1, W2, b2, W3, b3, out);
}